// SubspaceLinopFactory_53214644797865
// MI455X (gfx1250) — compile-verified
//
#include <hip/hip_runtime.h>
#include <hip/hip_bf16.h>

// SubspaceLinop NUDFT forward: y[t,c,k] = dcf * (2/96) * sum_p img[t,c,p]*exp(-i*om.r)
// Separable twiddles + part-reordered complex-rotation recurrence (no sincos,
// no cndmask in the hot loop), complex GEMM packed into f32 WMMA 16x16x4 tiles
// (M=16 = 2 parts x 8 coils, N=16 = 2 parts x 8 k-points, reduce over HW=9216).
// Hot loop: 1 global_load_b64 + 6 VALU + 1 v_wmma_f32_16x16x4_f32.

typedef __attribute__((ext_vector_type(2))) float v2f;
typedef __attribute__((ext_vector_type(8))) float v8f;

#define A_DIM 5
#define TT 32
#define CC 8
#define HH 96
#define WW 96
#define KK 3072
#define HW (HH * WW)
#define KTILES (KK / 8)  // 384 tiles of 8 complex k-points

// ---------------------------------------------------------------------------
// Prep: imgA[t][p/2][m][p%2], m = part*8 + c  (part 0 = re, 1 = im).
// Folds phi-expansion, coil sensitivity, and ortho scale 2/96.
// One float2 (b64) load per lane per WMMA in the main kernel.
// Size: 32 * 4608 * 16 * 2 floats = 18.9 MB in d_ws (L2-resident).
// ---------------------------------------------------------------------------
__global__ __launch_bounds__(256) void nudft_prep_kernel(
    const float* __restrict__ x, const float* __restrict__ phi,
    const float* __restrict__ mps, float* __restrict__ wsA) {
  int tid = blockIdx.x * 256 + threadIdx.x;
  if (tid >= TT * HW) return;
  int t = tid / HW;
  int p = tid - t * HW;

  float xtr = 0.f, xti = 0.f;
#pragma unroll
  for (int a = 0; a < A_DIM; ++a) {
    float pr = phi[(a * TT + t) * 2 + 0];
    float pi = phi[(a * TT + t) * 2 + 1];
    float xr = x[((size_t)a * HW + p) * 2 + 0];
    float xi = x[((size_t)a * HW + p) * 2 + 1];
    xtr += pr * xr - pi * xi;
    xti += pr * xi + pi * xr;
  }
  const float scale = 2.0f / 96.0f;  // 2^(D/2) / sqrt(H*W)
  xtr *= scale;
  xti *= scale;

  size_t base = ((size_t)t * (HW / 2) + (p >> 1)) * 32 + (p & 1);
#pragma unroll
  for (int c = 0; c < CC; ++c) {
    float mr = mps[((size_t)c * HW + p) * 2 + 0];
    float mi = mps[((size_t)c * HW + p) * 2 + 1];
    wsA[base + (size_t)c * 2]       = xtr * mr - xti * mi;  // re rows M=0..7
    wsA[base + (size_t)(8 + c) * 2] = xtr * mi + xti * mr;  // im rows M=8..15
  }
}

// ---------------------------------------------------------------------------
// Main: one wave per (t, k-tile); 2304 WMMAs over the 9216-pixel reduction.
// Twiddle e = exp(-i*phase) kept per-lane in part-reordered form (u, v):
//   u = the component this lane's B column consumes, v = the other one.
// Rotation by exp(-i*4*om_x) then needs only per-lane-constant coefficients
// (s4r, qa) -> zero cndmask; second B row derived as u1 = u*sxr + v*qs.
// ---------------------------------------------------------------------------
__global__ __launch_bounds__(256) void nudft_wmma_kernel(
    const float* __restrict__ trj, const float* __restrict__ sqrt_dcf,
    const float* __restrict__ wsA, float* __restrict__ out) {
  const int wv = threadIdx.x >> 5;     // wave in block (8 waves)
  const int lane = threadIdx.x & 31;
  const int w = blockIdx.x * 8 + wv;   // global wave id
  const int t = w / KTILES;            // all 8 waves of a block share t
  const int ktile = w - t * KTILES;
  const int kbase = ktile * 8;

  const int lh = lane >> 4;            // lane half: owns K-rows {2lh, 2lh+1}
  const int n = lane & 15;             // B/D column = 2*klocal + partb; A/D row
  const int klocal = n >> 1;
  const int partb = n & 1;             // 0: e_re column, 1: e_im column
  const int k = kbase + klocal;

  const float omy = trj[(t * 2 + 0) * KK + k];
  const float omx = trj[(t * 2 + 1) * KK + k];

  // ey = exp(-i*omy*(y-48)) at y=0  ->  exp(+i*48*omy)
  float eyr = __cosf(48.f * omy), eyi = __sinf(48.f * omy);
  // per-row step exp(-i*omy)
  const float syr = __cosf(omy), syi = -__sinf(omy);
  // ex at x=2*lh: exp(-i*omx*(2lh-48)) = exp(+i*omx*(48-2lh))
  const float axb = omx * (48.f - 2.f * (float)lh);
  const float exbr = __cosf(axb), exbi = __sinf(axb);
  // steps exp(-i*omx) and exp(-i*4*omx)
  const float sxr = __cosf(omx), sxi = -__sinf(omx);
  const float s4r = __cosf(4.f * omx), s4i = -__sinf(4.f * omx);
  // per-lane part-reordered coefficients (constants for the whole tile):
  //   re-lane: u=cr,v=ci:  u1 = u*sxr - v*sxi ; u' = u*s4r - v*s4i
  //   im-lane: u=ci,v=cr:  u1 = u*sxr + v*sxi ; u' = u*s4r + v*s4i
  const float qs = partb ? sxi : -sxi;
  const float qa = partb ? s4i : -s4i;

  const float2* __restrict__ Apairs = (const float2*)wsA;
  const size_t tbase = (size_t)t * (HW / 2) * 16;  // float2 units
  const int m = n;                                  // A row for this lane

  v8f acc = {};
  for (int y = 0; y < HH; ++y) {
    // full twiddle at (y, x = 2*lh), part-reordered into (u, v)
    const float c0r = eyr * exbr - eyi * exbi;
    const float c0i = eyr * exbi + eyi * exbr;
    float u = partb ? c0i : c0r;
    float v = partb ? c0r : c0i;

    const float2* rowp = Apairs + tbase + (size_t)(y * 48 + lh) * 16 + m;
    __builtin_prefetch(rowp + (size_t)48 * 16, 0, 0);  // next image row

#pragma unroll
    for (int j = 0; j < 24; ++j) {
      float2 a2 = rowp[(size_t)(2 * j) * 16];   // A[m][p], A[m][p+1]
      v2f av;
      av[0] = a2.x;
      av[1] = a2.y;
      v2f bv;                                   // B rows p, p+1 at column n
      bv[0] = u;
      bv[1] = u * sxr + v * qs;                 // twiddle at x+1, own part
      acc = __builtin_amdgcn_wmma_f32_16x16x4_f32(
          false, av, false, bv, (short)0, acc, false, false);
      // rotate (u,v) by exp(-i*4*omx) in reordered coordinates
      float nu = u * s4r + v * qa;
      float nv = v * s4r - u * qa;
      u = nu;
      v = nv;
    }
    // advance ey by exp(-i*omy)
    float ty = eyr * syr - eyi * syi;
    eyi = eyr * syi + eyi * syr;
    eyr = ty;
  }

  // Recombine partial products across lanes:
  //   lanes 0-15 hold part_a = re (M=0..7), lanes 16-31 part_a = im (M=8..15).
  //   out_re[c,k] = P_rr - P_ii ; out_im[c,k] = P_ri + P_ir.
  __shared__ float lds[8][32][8];
#pragma unroll
  for (int r = 0; r < 8; ++r) lds[wv][lane][r] = acc[r];
  __syncthreads();

  if (lane < 16) {
    const float dcf = sqrt_dcf[t * KK + k];
    const int isIm = partb;
    const int partner = isIm ? (16 + 2 * klocal) : (16 + 2 * klocal + 1);
    const float sgn = isIm ? 1.0f : -1.0f;
#pragma unroll
    for (int c = 0; c < CC; ++c) {
      float val = acc[c] + sgn * lds[wv][partner][c];
      out[(((size_t)t * CC + c) * KK + k) * 2 + isIm] = val * dcf;
    }
  }
}

// ---------------------------------------------------------------------------
extern "C" void kernel_launch(void* const* d_in, const int* in_sizes, int n_in,
                              void* d_out, int out_size, void* d_ws,
                              size_t ws_size, hipStream_t stream) {
  const float* x   = (const float*)d_in[0];  // [A,H,W] complex64 interleaved
  const float* phi = (const float*)d_in[1];  // [A,T]  complex64
  const float* mps = (const float*)d_in[2];  // [C,H,W] complex64
  const float* trj = (const float*)d_in[3];  // [T,D,K] float32
  const float* dcf = (const float*)d_in[4];  // [T,K]   float32
  // d_in[5] = subsamp_idx (identity, unused)
  float* wsA = (float*)d_ws;                 // needs 18.9 MB
  float* out = (float*)d_out;                // [T,C,K] complex64 interleaved

  nudft_prep_kernel<<<(TT * HW + 255) / 256, 256, 0, stream>>>(x, phi, mps, wsA);
  nudft_wmma_kernel<<<(TT * KTILES) / 8, 256, 0, stream>>>(trj, dcf, wsA, out);
}